// GraphSAGE_79834852098716
// MI455X (gfx1250) — compile-verified
//
#include <hip/hip_runtime.h>
#include <hip/hip_bf16.h>

// ---------------------------------------------------------------------------
// GraphSAGE (2x SAGEConv + BN + ReLU + linear head) for MI455X / gfx1250.
//   - scatter-add aggregation with global_atomic_add_f32 (no-return atomics),
//     float4 gathers, 4 features/thread to minimize index traffic
//   - GEMMs via V_WMMA_F32_16X16X4_F32 (exact f32, matches f32 reference);
//     4 N-tiles per wave: A fragments reused, 4 independent accumulator chains
//   - mean-normalization, bias, BatchNorm(eval), ReLU fused into GEMM epilogue
// ---------------------------------------------------------------------------

#define N_NODES 50000
#define N_EDGES 600000
#define D 128
#define BN_EPS 1e-5f
#define NT 4   // N-tiles (16 cols each) per wave

typedef __attribute__((ext_vector_type(2))) float v2f;
typedef __attribute__((ext_vector_type(8))) float v8f;

// ---------------------------------------------------------------- zero fill
__global__ __launch_bounds__(256) void zero_kernel(float* __restrict__ p, int n) {
    int i = blockIdx.x * 256 + threadIdx.x;
    if (i < n) p[i] = 0.0f;
}

// ------------------------------------------------------------- scatter-add
// One thread per (edge, 4-feature chunk): 600000*32 = 19.2M threads.
// float4 gather from x[src], 4 scalar f32 atomics into agg[dst].
template <bool COUNT>
__global__ __launch_bounds__(256) void scatter_kernel(
    const float* __restrict__ x, const int* __restrict__ src,
    const int* __restrict__ dst, float* __restrict__ agg,
    float* __restrict__ cnt) {
    int t = blockIdx.x * 256 + threadIdx.x;   // < 19.2M
    int e = t >> 5;                            // edge id
    int q = (t & 31) << 2;                     // feature base: 0,4,...,124
    if (e >= N_EDGES) return;
    int s = src[e];
    int d = dst[e];
    const float4 v = *(const float4*)(x + (size_t)s * D + q);
    float* a = agg + (size_t)d * D + q;
    atomicAdd(a + 0, v.x);
    atomicAdd(a + 1, v.y);
    atomicAdd(a + 2, v.z);
    atomicAdd(a + 3, v.w);
    if (COUNT && (t & 31) == 0) atomicAdd(&cnt[d], 1.0f);
}

// ------------------------------------------------------------ fused SAGE GEMM
// out[m][n] = act( (agg[m]/max(cnt,1)) @ Wl.T + bl + xin[m] @ Wr.T ), opt. BN.
// One wave handles a 16-row slab x 4 N-tiles (64 cols). K steps of 4 via
// V_WMMA_F32_16X16X4_F32; A fragments loaded once per k-step, reused 4x.
//
// f32 WMMA fragment layouts (ISA 7.12.2):
//   A 16x4:  lane L<16 holds A[L][k], A[L][k+1];  lane L+16 holds A[L][k+2..3]
//   B 4x16:  lane L<16 holds B[k][L], B[k+1][L];  lane L+16 holds B[k+2..3][L]
//   C/D 16x16: VGPR j -> row j (lanes 0-15) / row j+8 (lanes 16-31), col=lane&15
template <bool BN>
__global__ __launch_bounds__(32) void sage_gemm_kernel(
    const float* __restrict__ agg, const float* __restrict__ cnt,
    const float* __restrict__ xin, const float* __restrict__ Wl,
    const float* __restrict__ Wr, const float* __restrict__ bl,
    const float* __restrict__ bn_gamma, const float* __restrict__ bn_beta,
    const float* __restrict__ bn_mean, const float* __restrict__ bn_var,
    float* __restrict__ out) {
    const int lane = threadIdx.x;                 // 0..31, wave32
    const int lm   = lane & 15;
    const int m0   = blockIdx.x * 16;             // 3125 * 16 == 50000 exact
    const int n0   = blockIdx.y * (NT * 16);      // 0 or 64
    const int kOff = (lane >> 4) * 2;             // K sub-offset 0 or 2
    const int rowA = m0 + lm;

    // mean normalization factor for the agg operand (fused)
    const float rinv = 1.0f / fmaxf(cnt[rowA], 1.0f);

    const float* aggRow = agg + rowA * D + kOff;
    const float* xRow   = xin + rowA * D + kOff;
    const float* wlBase = Wl + (n0 + lm) * D + kOff;  // B1[k][n] = Wl[n][k]
    const float* wrBase = Wr + (n0 + lm) * D + kOff;  // B2[k][n] = Wr[n][k]

    v8f c[NT] = {};

    for (int k = 0; k < D; k += 4) {
        v2f a1 = *(const v2f*)(aggRow + k);
        a1.x *= rinv;
        a1.y *= rinv;
        v2f a2 = *(const v2f*)(xRow + k);
#pragma unroll
        for (int t = 0; t < NT; ++t) {
            v2f b1 = *(const v2f*)(wlBase + t * 16 * D + k);
            c[t] = __builtin_amdgcn_wmma_f32_16x16x4_f32(false, a1, false, b1,
                                                         (short)0, c[t], false,
                                                         false);
        }
#pragma unroll
        for (int t = 0; t < NT; ++t) {
            v2f b2 = *(const v2f*)(wrBase + t * 16 * D + k);
            c[t] = __builtin_amdgcn_wmma_f32_16x16x4_f32(false, a2, false, b2,
                                                         (short)0, c[t], false,
                                                         false);
        }
    }

    // ------------------------------------------------------------ epilogue
    const int rbase = m0 + ((lane >> 4) << 3);    // +0 or +8
#pragma unroll
    for (int t = 0; t < NT; ++t) {
        const int nc = n0 + t * 16 + lm;
        const float bias = bl[nc];
        float scale = 1.0f, shift = 0.0f;
        if constexpr (BN) {
            scale = bn_gamma[nc] * rsqrtf(bn_var[nc] + BN_EPS);
            shift = bn_beta[nc] - bn_mean[nc] * scale;
        }
#pragma unroll
        for (int j = 0; j < 8; ++j) {
            float v = c[t][j] + bias;
            if constexpr (BN) v = v * scale + shift;
            v = fmaxf(v, 0.0f);                   // ReLU (dropout = identity)
            out[(rbase + j) * D + nc] = v;
        }
    }
}

// ----------------------------------------------------------------- head GEMV
// logits[m] = relu(h2[m]) @ Wc.T + bc ; relu already applied in gemm epilogue.
__global__ __launch_bounds__(32) void logits_kernel(
    const float* __restrict__ h2, const float* __restrict__ Wc,
    const float* __restrict__ bc, float* __restrict__ outv) {
    int node = blockIdx.x;
    int lane = threadIdx.x;
    float s = 0.0f;
#pragma unroll
    for (int k = lane; k < D; k += 32) s += h2[node * D + k] * Wc[k];
#pragma unroll
    for (int off = 16; off > 0; off >>= 1) s += __shfl_xor(s, off, 32);
    if (lane == 0) outv[node] = s + bc[0];
}

// ---------------------------------------------------------------------------
extern "C" void kernel_launch(void* const* d_in, const int* in_sizes, int n_in,
                              void* d_out, int out_size, void* d_ws,
                              size_t ws_size, hipStream_t stream) {
    const float* x        = (const float*)d_in[0];
    const int*   ei       = (const int*)d_in[1];
    const float* Wl1      = (const float*)d_in[2];
    const float* bl1      = (const float*)d_in[3];
    const float* Wr1      = (const float*)d_in[4];
    const float* bn_gamma = (const float*)d_in[5];
    const float* bn_beta  = (const float*)d_in[6];
    const float* bn_mean  = (const float*)d_in[7];
    const float* bn_var   = (const float*)d_in[8];
    const float* Wl2      = (const float*)d_in[9];
    const float* bl2      = (const float*)d_in[10];
    const float* Wr2      = (const float*)d_in[11];
    const float* Wc       = (const float*)d_in[12];
    const float* bc       = (const float*)d_in[13];

    const int* src = ei;
    const int* dst = ei + N_EDGES;

    // workspace layout (floats): [agg | cnt | h | h2]  ~77 MB total
    float* ws  = (float*)d_ws;
    float* agg = ws;                               // N_NODES * D
    float* cnt = agg + (size_t)N_NODES * D;        // N_NODES
    float* h   = cnt + N_NODES;                    // N_NODES * D
    float* h2  = h + (size_t)N_NODES * D;          // N_NODES * D

    const int nAgg = N_NODES * D;                  // 6,400,000
    const int nScatterBlocks = (N_EDGES * 32) / 256;  // 75,000 exact
    dim3 gemmGrid(N_NODES / 16, D / (NT * 16));    // 3125 x 2

    // ---------------- layer 1 ----------------
    zero_kernel<<<(nAgg + N_NODES + 255) / 256, 256, 0, stream>>>(agg,
                                                                  nAgg + N_NODES);
    scatter_kernel<true><<<nScatterBlocks, 256, 0, stream>>>(x, src, dst, agg,
                                                             cnt);
    sage_gemm_kernel<true><<<gemmGrid, 32, 0, stream>>>(
        agg, cnt, x, Wl1, Wr1, bl1, bn_gamma, bn_beta, bn_mean, bn_var, h);

    // ---------------- layer 2 ----------------
    zero_kernel<<<(nAgg + 255) / 256, 256, 0, stream>>>(agg, nAgg);
    scatter_kernel<false><<<nScatterBlocks, 256, 0, stream>>>(h, src, dst, agg,
                                                              cnt);
    sage_gemm_kernel<false><<<gemmGrid, 32, 0, stream>>>(
        agg, cnt, h, Wl2, Wr2, bl2, nullptr, nullptr, nullptr, nullptr, h2);

    // ---------------- head ----------------
    logits_kernel<<<N_NODES, 32, 0, stream>>>(h2, Wc, bc, (float*)d_out);
}